// SingleRoIExtractor3DChimp_55860344652109
// MI455X (gfx1250) — compile-verified
//
#include <hip/hip_runtime.h>

typedef __attribute__((ext_vector_type(2))) float v2f;
typedef __attribute__((ext_vector_type(8))) float v8f;

#define ROI_OUT 16
#define N_C 256
#define N_T 16
#define N_K 5
#define N_H 56
#define N_W 56
#define ROI_SCALE (1.0f/16.0f)

// out = A(16x56) * feat(56x56) * B^T(56x16) per (n,k,t,c) map, via V_WMMA_F32_16X16X4_F32.
__global__ __launch_bounds__(256) void roialign_wmma(const float* __restrict__ feat,
                                                     const float* __restrict__ rois,
                                                     float* __restrict__ out)
{
    __shared__ float Awt[16][57];        // y-axis weights, padded to kill bank conflicts
    __shared__ float Bwt[16][57];        // x-axis weights
    __shared__ float Smat[8][64][16];    // per-wave intermediate S = feat * B^T

    int b = blockIdx.x;
    int cblk = b & 1;  b >>= 1;
    int t    = b & 15; b >>= 4;
    int k    = b % N_K; b /= N_K;
    int n    = b;

    int tid = threadIdx.x;

    // ---- Build separable weight matrices A (y) and B (x) in LDS ----
    if (tid < 32) {
        int axis = tid >> 4;     // 0 -> A (y axis, rows = py), 1 -> B (x axis, rows = px)
        int p    = tid & 15;
        const float* r = rois + (n * N_K + k) * 5;   // [batch, x1, y1, x2, y2]
        float lo  = (axis == 0 ? r[2] : r[1]) * ROI_SCALE - 0.5f;
        float hi  = (axis == 0 ? r[4] : r[3]) * ROI_SCALE - 0.5f;
        float bin = (hi - lo) * (1.0f / ROI_OUT);
        float* row = (axis == 0 ? &Awt[p][0] : &Bwt[p][0]);
        for (int i = 0; i < N_W; ++i) row[i] = 0.0f;
        const float lim = (float)N_H;    // H == W == 56
        #pragma unroll
        for (int g = 0; g < 2; ++g) {
            float pos = lo + ((float)p + ((float)g + 0.5f) * 0.5f) * bin;
            bool valid = (pos > -1.0f) && (pos < lim);
            float pc = fminf(fmaxf(pos, 0.0f), lim - 1.0f);
            int i0 = (int)floorf(pc);
            int i1 = min(i0 + 1, N_H - 1);
            float l = pc - (float)i0;
            if (valid) { row[i0] += 0.5f * (1.0f - l); row[i1] += 0.5f * l; }
        }
    }
    __syncthreads();

    int wave = tid >> 5;
    int lane = tid & 31;
    int m15  = lane & 15;
    int hig  = lane >> 4;                 // lane half: 0 or 1
    float (*S)[16] = Smat[wave];

    for (int ci = 0; ci < 16; ++ci) {
        int c = cblk * 128 + wave * 16 + ci;
        const float* fm = feat + (((size_t)(n * N_C + c) * N_T + t) * (N_H * N_W));

        // ---- Stage 1: S(64x16) = feat(56x56, zero-pad to 64) * B^T(56x16) ----
        #pragma unroll
        for (int mt = 0; mt < 4; ++mt) {
            v8f acc = {};
            #pragma unroll
            for (int cc = 0; cc < 14; ++cc) {
                int row = mt * 16 + m15;
                int col = 4 * cc + hig * 2;
                v2f a;
                if (row < N_H) {                       // only mt==3 can pad
                    const float* p = fm + row * N_W + col;
                    a.x = p[0]; a.y = p[1];
                } else { a.x = 0.0f; a.y = 0.0f; }
                v2f bv;
                bv.x = Bwt[m15][col];
                bv.y = Bwt[m15][col + 1];
                acc = __builtin_amdgcn_wmma_f32_16x16x4_f32(
                    false, a, false, bv, (short)0, acc, false, false);
            }
            // D (C/D layout) -> LDS S[y][px]
            #pragma unroll
            for (int v = 0; v < 8; ++v) {
                S[mt * 16 + v + hig * 8][m15] = acc[v];
            }
        }

        // ---- Stage 2: out(16x16) = A(16x56) * S(56x16) ----
        v8f od = {};
        #pragma unroll
        for (int cc = 0; cc < 14; ++cc) {
            int col = 4 * cc + hig * 2;
            v2f a;
            a.x = Awt[m15][col];
            a.y = Awt[m15][col + 1];
            v2f bv;
            bv.x = S[col][m15];
            bv.y = S[col + 1][m15];
            od = __builtin_amdgcn_wmma_f32_16x16x4_f32(
                false, a, false, bv, (short)0, od, false, false);
        }

        size_t base = ((((size_t)((n * N_T + t) * N_K + k)) * N_C + c) * (ROI_OUT * ROI_OUT));
        #pragma unroll
        for (int v = 0; v < 8; ++v) {
            int py = v + hig * 8;
            out[base + py * ROI_OUT + m15] = od[v];
        }
    }
}

// feat passthrough: wide 128-bit grid-stride copy (pure-bandwidth path).
__global__ void copy_feat(const float4* __restrict__ src, float4* __restrict__ dst, long n4)
{
    long i = (long)blockIdx.x * blockDim.x + threadIdx.x;
    long stride = (long)gridDim.x * blockDim.x;
    for (; i < n4; i += stride) dst[i] = src[i];
}

extern "C" void kernel_launch(void* const* d_in, const int* in_sizes, int n_in,
                              void* d_out, int out_size, void* d_ws, size_t ws_size,
                              hipStream_t stream) {
    const float* feat = (const float*)d_in[0];
    const float* rois = (const float*)d_in[1];
    // d_in[2] (entity_mask) is unused by the reference outputs.
    float* out = (float*)d_out;

    const long roiElems = 2L * N_T * N_K * N_C * ROI_OUT * ROI_OUT;  // 10,485,760

    // 2 (n) * 5 (k) * 16 (t) * 2 (c-half) workgroups, 256 threads (8 wave32) each.
    roialign_wmma<<<320, 256, 0, stream>>>(feat, rois, out);

    long n4 = (long)in_sizes[0] / 4;   // 25,690,112 floats -> 6,422,528 float4
    copy_feat<<<2048, 256, 0, stream>>>((const float4*)feat, (float4*)(out + roiElems), n4);
}